// GraphPoolingOut_28587302322979
// MI455X (gfx1250) — compile-verified
//
#include <hip/hip_runtime.h>
#include <float.h>

typedef float v2f __attribute__((ext_vector_type(2)));
typedef float v4f __attribute__((ext_vector_type(4)));

#define NTHREADS 256

// ---- CDNA5 float-number atomics (non-returning, device scope) ----
__device__ __forceinline__ void atomic_max_f32_dev(float* addr, float v) {
    asm volatile("global_atomic_max_num_f32 %0, %1, off scope:SCOPE_DEV"
                 :: "v"(addr), "v"(v) : "memory");
}
__device__ __forceinline__ void atomic_min_f32_dev(float* addr, float v) {
    asm volatile("global_atomic_min_num_f32 %0, %1, off scope:SCOPE_DEV"
                 :: "v"(addr), "v"(v) : "memory");
}

// ---- CDNA5 async global->LDS copy (ASYNCcnt path) ----
// 16B per lane. DScnt and ASYNCcnt are mutually unordered, so our prior
// ds_loads from the target buffer must complete before the async engine may
// overwrite it: fuse s_wait_dscnt 0 into the issue.
__device__ __forceinline__ void async_copy_b128(unsigned lds_off, const void* gaddr) {
    asm volatile("s_wait_dscnt 0x0\n\t"
                 "global_load_async_to_lds_b128 %0, %1, off th:TH_LOAD_NT"
                 :: "v"(lds_off), "v"(gaddr) : "memory");
}
__device__ __forceinline__ void wait_async_le1() {
    asm volatile("s_wait_asynccnt 0x1" ::: "memory");
}
__device__ __forceinline__ void wait_async_0() {
    asm volatile("s_wait_asynccnt 0x0" ::: "memory");
}

// Low 32 bits of a generic pointer to __shared__ = workgroup-relative LDS
// byte offset (ISA 10.2: LDS aperture address truncates to addr[31:0]).
__device__ __forceinline__ unsigned lds_offset(const void* p) {
    return (unsigned)(unsigned long long)p;
}

// Kernel 1: sentinel-init workspace (min/max accumulators) and output (-FLT_MAX).
__global__ void k_init(float* __restrict__ out, int out_n, float* __restrict__ ws) {
    int i = blockIdx.x * blockDim.x + threadIdx.x;
    if (i == 0) {
        ws[0] =  FLT_MAX;  // min x
        ws[1] =  FLT_MAX;  // min y
        ws[2] = -FLT_MAX;  // max x
        ws[3] = -FLT_MAX;  // max y
    }
    int stride = gridDim.x * blockDim.x;
    for (int j = i; j < out_n; j += stride)
        out[j] = -FLT_MAX;
}

// Kernel 2: global min/max of pos. Bulk is an async-to-LDS double-buffered
// stream (2 points = 16B per lane per iteration); tail uses plain NT loads.
__global__ void k_minmax(const float* __restrict__ pos, int n, float* __restrict__ ws) {
    __shared__ v4f buf[2][NTHREADS];   // per-lane 16B slots, double buffered

    float mnx = FLT_MAX, mny = FLT_MAX, mxx = -FLT_MAX, mxy = -FLT_MAX;

    const int tid      = blockIdx.x * blockDim.x + threadIdx.x;
    const int nthr     = gridDim.x * blockDim.x;
    const int pts_iter = nthr * 2;            // 2 points per thread per iteration
    const int iters    = n / pts_iter;        // bulk via async pipeline

    const unsigned lds0 = lds_offset(&buf[0][threadIdx.x]);
    const unsigned lds1 = lds_offset(&buf[1][threadIdx.x]);

    if (iters > 0)
        async_copy_b128(lds0, pos + (long)tid * 4);        // prefetch chunk 0

    for (int k = 0; k < iters; ++k) {
        if (k + 1 < iters) {
            // prefetch chunk k+1 into the other buffer, then wait for chunk k
            async_copy_b128(((k + 1) & 1) ? lds1 : lds0,
                            pos + ((long)(k + 1) * pts_iter + (long)tid * 2) * 2);
            wait_async_le1();
        } else {
            wait_async_0();
        }
        v4f q = buf[k & 1][threadIdx.x];                   // ds_load_b128
        mnx = fminf(mnx, fminf(q.x, q.z));
        mny = fminf(mny, fminf(q.y, q.w));
        mxx = fmaxf(mxx, fmaxf(q.x, q.z));
        mxy = fmaxf(mxy, fmaxf(q.y, q.w));
    }

    // tail (points not covered by the pipeline)
    for (int i = iters * pts_iter + tid; i < n; i += nthr) {
        v2f p = __builtin_nontemporal_load((const v2f*)pos + i);
        mnx = fminf(mnx, p.x);
        mny = fminf(mny, p.y);
        mxx = fmaxf(mxx, p.x);
        mxy = fmaxf(mxy, p.y);
    }

    // wave32 butterfly reduce
    #pragma unroll
    for (int off = 16; off > 0; off >>= 1) {
        mnx = fminf(mnx, __shfl_xor(mnx, off));
        mny = fminf(mny, __shfl_xor(mny, off));
        mxx = fmaxf(mxx, __shfl_xor(mxx, off));
        mxy = fmaxf(mxy, __shfl_xor(mxy, off));
    }
    __shared__ float s[4][NTHREADS / 32];
    int wave = threadIdx.x >> 5;
    int lane = threadIdx.x & 31;
    if (lane == 0) {
        s[0][wave] = mnx; s[1][wave] = mny; s[2][wave] = mxx; s[3][wave] = mxy;
    }
    __syncthreads();
    if (threadIdx.x == 0) {
        #pragma unroll
        for (int w = 1; w < NTHREADS / 32; ++w) {
            mnx = fminf(mnx, s[0][w]);
            mny = fminf(mny, s[1][w]);
            mxx = fmaxf(mxx, s[2][w]);
            mxy = fmaxf(mxy, s[3][w]);
        }
        atomic_min_f32_dev(&ws[0], mnx);
        atomic_min_f32_dev(&ws[1], mny);
        atomic_max_f32_dev(&ws[2], mxx);
        atomic_max_f32_dev(&ws[3], mxy);
    }
}

// Kernel 3: scatter-max. 16 consecutive lanes handle the 16 features of one
// point, so x loads are fully coalesced; one GLOBAL_ATOMIC_MAX_NUM_F32 per
// feature element, device scope, no return (STOREcnt only).
__global__ void k_scatter(const float* __restrict__ pos,
                          const float* __restrict__ x,
                          const int*   __restrict__ batch,
                          const float* __restrict__ ws,
                          float* __restrict__ out,
                          int n, int out_rows) {
    const float sx = ws[0], sy = ws[1], ex = ws[2];
    const int nx = (int)(floorf((ex - sx) * 0.25f) + 1.0f);   // VOXEL = 4.0 exact

    const long total  = (long)n * 16;
    const long stride = (long)gridDim.x * blockDim.x;
    for (long i = (long)blockIdx.x * blockDim.x + threadIdx.x; i < total; i += stride) {
        const int point = (int)(i >> 4);
        const int feat  = (int)(i & 15);
        // 16 lanes share the same address -> single coalesced request per wave
        const float px = pos[2 * point];
        const float py = pos[2 * point + 1];
        const int   b  = batch[point];
        const int cx = (int)floorf((px - sx) * 0.25f);
        const int cy = (int)floorf((py - sy) * 0.25f);
        const int cluster = cx + cy * nx + b * 5655;
        if (cluster >= 0 && cluster < out_rows) {             // mode='drop'
            const float v = __builtin_nontemporal_load(x + i);
            atomic_max_f32_dev(out + (long)cluster * 16 + feat, v);
        }
    }
}

// Kernel 4: untouched cells (-FLT_MAX sentinel) -> 0.
__global__ void k_final(float* __restrict__ out, int n) {
    int i = blockIdx.x * blockDim.x + threadIdx.x;
    if (i < n) {
        float v = out[i];
        out[i] = (v == -FLT_MAX) ? 0.0f : v;
    }
}

extern "C" void kernel_launch(void* const* d_in, const int* in_sizes, int n_in,
                              void* d_out, int out_size, void* d_ws, size_t ws_size,
                              hipStream_t stream) {
    const float* pos   = (const float*)d_in[0];
    const float* x     = (const float*)d_in[1];
    const int*   batch = (const int*)d_in[2];
    float* out = (float*)d_out;
    float* ws  = (float*)d_ws;

    const int n        = in_sizes[2];      // number of points
    const int out_rows = out_size / 16;    // B * SIZE

    k_init  <<<1024, NTHREADS, 0, stream>>>(out, out_size, ws);
    k_minmax<<<1024, NTHREADS, 0, stream>>>(pos, n, ws);

    long total = (long)n * 16;
    long want  = (total + NTHREADS - 1) / NTHREADS;
    int blocks = (int)(want < 16384 ? want : 16384);
    if (blocks < 1) blocks = 1;
    k_scatter<<<blocks, NTHREADS, 0, stream>>>(pos, x, batch, ws, out, n, out_rows);

    k_final <<<(out_size + NTHREADS - 1) / NTHREADS, NTHREADS, 0, stream>>>(out, out_size);
}